// topk_33586644255176
// MI455X (gfx1250) — compile-verified
//
#include <hip/hip_runtime.h>
#include <math.h>

typedef float v2f __attribute__((ext_vector_type(2)));
typedef float v8f __attribute__((ext_vector_type(8)));

#define G_      128
#define NPER_   512
#define NTOT    65536
#define ETOT    1048576
#define EPG     8192
#define IN_     256
#define D_      128
#define OUT_    16
#define K1_     256
#define K2_     128
#define N1_     (G_ * K1_)   // 32768
#define N2_     (G_ * K2_)   // 16384

// d_out layout (floats): [0,2048) logits, [2048,34816) g_emb,
// [34816,34944) pool1_w, [34944,67712) score_1
#define OFF_GEMB  2048
#define OFF_P1W   34816
#define OFF_SCORE 34944

// ---------------------------------------------------------------------------
// Fused dual GEMM: C0 = A@W0 ; C1 = A@W1 + bias.  A:[rows,K], W:[K,128].
// Block = 256 threads (8 waves). Block tile = 32 rows x 128 cols; each wave
// owns one 16-col strip and TWO 16-row subtiles that share B fragments, so
// each k-step issues 4 v_wmma_f32_16x16x4_f32 per 4 B loads + 2 LDS loads.
// ---------------------------------------------------------------------------
template <int K>
__global__ void dual_gemm_kernel(const float* __restrict__ A,
                                 const float* __restrict__ W0,
                                 const float* __restrict__ W1,
                                 const float* __restrict__ bias,
                                 float* __restrict__ C0,
                                 float* __restrict__ C1) {
  __shared__ float As[32 * (K + 2)];
  const int tid      = threadIdx.x;
  const int rowBlock = blockIdx.x * 32;

  // Cooperative, coalesced A-tile load into LDS (padded row stride K+2).
  for (int i = tid; i < 32 * K; i += 256) {
    const int r = i / K, k = i % K;
    As[r * (K + 2) + k] = A[(rowBlock + r) * K + k];
  }
  __syncthreads();

  const int wave = tid >> 5;                 // 0..7 -> 16-col strip
  const int lane = tid & 31;
  const int hr   = lane & 15;                // row-in-tile (A) / col (B,C,D)
  const int hi   = lane >> 4;                // half-wave select
  const int n    = wave * 16 + hr;           // output column 0..127

  v8f c00 = {}; v8f c01 = {}; v8f c10 = {}; v8f c11 = {};
  const float* Ar0 = &As[hr        * (K + 2) + 2 * hi];   // rows  0..15
  const float* Ar1 = &As[(16 + hr) * (K + 2) + 2 * hi];   // rows 16..31

  for (int k0 = 0; k0 < K; k0 += 4) {
    // A 16x4 fragments: lanes0-15 -> (K=k0,k0+1), lanes16-31 -> (k0+2,k0+3)
    v2f a0 = *(const v2f*)(Ar0 + k0);        // ds_load_b64
    v2f a1 = *(const v2f*)(Ar1 + k0);
    const int kk = k0 + 2 * hi;
    v2f b0, b1;
    b0.x = W0[kk * D_ + n]; b0.y = W0[(kk + 1) * D_ + n];
    b1.x = W1[kk * D_ + n]; b1.y = W1[(kk + 1) * D_ + n];
    c00 = __builtin_amdgcn_wmma_f32_16x16x4_f32(false, a0, false, b0,
                                                (short)0, c00, false, false);
    c01 = __builtin_amdgcn_wmma_f32_16x16x4_f32(false, a0, false, b1,
                                                (short)0, c01, false, false);
    c10 = __builtin_amdgcn_wmma_f32_16x16x4_f32(false, a1, false, b0,
                                                (short)0, c10, false, false);
    c11 = __builtin_amdgcn_wmma_f32_16x16x4_f32(false, a1, false, b1,
                                                (short)0, c11, false, false);
  }

  const float bv = bias[n];
  const int r0 = rowBlock + hi * 8;          // C/D: VGPR r -> M=r (+8 for hi)
  for (int r = 0; r < 8; ++r) {
    C0[(r0 + r) * D_ + n]      = c00[r];
    C1[(r0 + r) * D_ + n]      = c01[r] + bv;
    C0[(r0 + 16 + r) * D_ + n] = c10[r];
    C1[(r0 + 16 + r) * D_ + n] = c11[r] + bv;
  }
}

// ---------------------------------------------------------------------------
// Per-graph edge aggregation with whole-graph accumulator resident in LDS
// (CDNA5: 320KB/WGP). hacc init = x@w_root + b_rel; edges scatter xr[src]
// into hacc[dst] via ds_add_f32; then ReLU to global.
// ---------------------------------------------------------------------------
template <int NPOOL, bool USE_MAP>
__global__ void agg_relu_kernel(const float* __restrict__ xr,
                                const float* __restrict__ accin,
                                float* __restrict__ hout,
                                const int* __restrict__ edge_index,
                                const int* __restrict__ mapping) {
  extern __shared__ float hacc[];                // NPOOL*128 floats
  const int g   = blockIdx.x;
  const int tid = threadIdx.x;                   // 512
  const int FEATS = NPOOL * D_;

  for (int i = tid; i < FEATS; i += 512) hacc[i] = accin[g * FEATS + i];
  __syncthreads();

  const int egrp  = tid >> 7;                    // 4 edges in flight
  const int f     = tid & 127;
  const int ebase = g * EPG;
  for (int e = egrp; e < EPG; e += 4) {
    // stream-ahead prefetch of the edge index arrays (global_prefetch_b8)
    __builtin_prefetch(&edge_index[ebase + e + 64], 0, 0);
    __builtin_prefetch(&edge_index[ETOT + ebase + e + 64], 0, 0);
    int s = edge_index[ebase + e];
    int d = edge_index[ETOT + ebase + e];
    if (USE_MAP) {
      s = mapping[s]; d = mapping[d];
      if (s < 0 || d < 0) continue;
    }
    const int dl = d - g * NPOOL;
    atomicAdd(&hacc[dl * D_ + f], xr[s * D_ + f]);   // ds_add_f32
  }
  __syncthreads();

  for (int i = tid; i < FEATS; i += 512)
    hout[g * FEATS + i] = fmaxf(hacc[i], 0.0f);
}

// ---------------------------------------------------------------------------
// TopK pooling: score = sigmoid(h.w/||w||); bitonic sort (desc score, asc idx
// tie-break == lax.top_k); emit gathered x_new = h[perm]*score, relabel map,
// optional score output.
// ---------------------------------------------------------------------------
template <int NPER, int KKEEP>
__global__ void topk_pool_kernel(const float* __restrict__ h,
                                 const float* __restrict__ w,
                                 float* __restrict__ xnew,
                                 int* __restrict__ mapping,
                                 float* __restrict__ score_out) {
  __shared__ float sc[NPER];
  __shared__ int   si[NPER];
  const int g = blockIdx.x, t = threadIdx.x;

  float s2 = 0.0f;
  for (int i = 0; i < D_; ++i) { const float wv = w[i]; s2 += wv * wv; }
  const float invn = 1.0f / sqrtf(s2);

  const int node = g * NPER + t;
  float dot = 0.0f;
  for (int f = 0; f < D_; ++f) dot += h[node * D_ + f] * w[f];
  sc[t] = 1.0f / (1.0f + __expf(-dot * invn));
  si[t] = t;
  mapping[node] = -1;

  // Bitonic sort, descending by (score, -idx).
  for (int size = 2; size <= NPER; size <<= 1) {
    for (int stride = size >> 1; stride > 0; stride >>= 1) {
      __syncthreads();
      const int p = t ^ stride;
      if (p > t) {
        const float a = sc[t], b = sc[p];
        const int ia = si[t], ib = si[p];
        const bool aFirst = (a > b) || (a == b && ia < ib);
        const bool desc   = ((t & size) == 0);
        if (desc ? !aFirst : aFirst) {
          sc[t] = b; sc[p] = a; si[t] = ib; si[p] = ia;
        }
      }
    }
  }
  __syncthreads();

  if (t < KKEEP) {
    mapping[g * NPER + si[t]] = g * KKEEP + t;
    if (score_out) score_out[g * KKEEP + t] = sc[t];
  }
  for (int i = t; i < KKEEP * D_; i += NPER) {
    const int r = i >> 7, f = i & 127;
    xnew[(g * KKEEP + r) * D_ + f] = h[(g * NPER + si[r]) * D_ + f] * sc[r];
  }
}

// ---------------------------------------------------------------------------
// Per-graph global max+mean pool -> gx[g] = [max(128) | mean(128)]
// ---------------------------------------------------------------------------
__global__ void global_pool_kernel(const float* __restrict__ xnew,
                                   float* __restrict__ gx, int kk) {
  const int g = blockIdx.x, f = threadIdx.x;     // 128 threads
  const float* base = xnew + g * kk * D_ + f;
  float m = -INFINITY, s = 0.0f;
  for (int r = 0; r < kk; ++r) {
    const float v = base[r * D_];
    m = fmaxf(m, v); s += v;
  }
  gx[g * 256 + f]       = m;
  gx[g * 256 + 128 + f] = s / (float)kk;
}

// ---------------------------------------------------------------------------
// Head: g_emb = gx1+gx2; z = BN(relu(g_emb@lin1+b)); out = logsoftmax(z@lin2+b)
// Also emits g_emb and pool1_w copy into d_out.
// ---------------------------------------------------------------------------
__global__ void head_kernel(const float* __restrict__ gx1,
                            const float* __restrict__ gx2,
                            const float* __restrict__ lin1_w,
                            const float* __restrict__ lin1_b,
                            const float* __restrict__ bng,
                            const float* __restrict__ bnb,
                            const float* __restrict__ bnm,
                            const float* __restrict__ bnv,
                            const float* __restrict__ lin2_w,
                            const float* __restrict__ lin2_b,
                            const float* __restrict__ pool1_w,
                            float* __restrict__ out) {
  __shared__ float ge[2 * D_];
  __shared__ float z[D_];
  __shared__ float logits[OUT_];
  __shared__ float red[2];
  const int g = blockIdx.x, t = threadIdx.x;     // 128 threads

  for (int i = t; i < 2 * D_; i += D_) {
    const float v = gx1[g * 256 + i] + gx2[g * 256 + i];
    ge[i] = v;
    out[OFF_GEMB + g * 256 + i] = v;
  }
  if (g == 0) out[OFF_P1W + t] = pool1_w[t];
  __syncthreads();

  float acc = lin1_b[t];
  for (int i = 0; i < 2 * D_; ++i) acc += ge[i] * lin1_w[i * D_ + t];
  acc = fmaxf(acc, 0.0f);
  acc = (acc - bnm[t]) * bng[t] / sqrtf(bnv[t] + 1e-5f) + bnb[t];
  z[t] = acc;
  __syncthreads();

  if (t < OUT_) {
    float l = lin2_b[t];
    for (int j = 0; j < D_; ++j) l += z[j] * lin2_w[j * OUT_ + t];
    logits[t] = l;
  }
  __syncthreads();
  if (t == 0) {
    float m = logits[0];
    for (int i = 1; i < OUT_; ++i) m = fmaxf(m, logits[i]);
    float se = 0.0f;
    for (int i = 0; i < OUT_; ++i) se += __expf(logits[i] - m);
    red[0] = m; red[1] = logf(se);
  }
  __syncthreads();
  if (t < OUT_) out[g * OUT_ + t] = logits[t] - red[0] - red[1];
}

// ---------------------------------------------------------------------------
extern "C" void kernel_launch(void* const* d_in, const int* in_sizes, int n_in,
                              void* d_out, int out_size, void* d_ws, size_t ws_size,
                              hipStream_t stream) {
  (void)in_sizes; (void)n_in; (void)out_size; (void)ws_size;

  const float* x        = (const float*)d_in[0];
  const int*   ei       = (const int*)d_in[1];
  const float* w_rel1   = (const float*)d_in[3];
  const float* b_rel1   = (const float*)d_in[4];
  const float* w_root1  = (const float*)d_in[5];
  const float* pool1_w  = (const float*)d_in[6];
  const float* w_rel2   = (const float*)d_in[7];
  const float* b_rel2   = (const float*)d_in[8];
  const float* w_root2  = (const float*)d_in[9];
  const float* pool2_w  = (const float*)d_in[10];
  const float* lin1_w   = (const float*)d_in[11];
  const float* lin1_b   = (const float*)d_in[12];
  const float* bn_gamma = (const float*)d_in[13];
  const float* bn_beta  = (const float*)d_in[14];
  const float* bn_mean  = (const float*)d_in[15];
  const float* bn_var   = (const float*)d_in[16];
  const float* lin2_w   = (const float*)d_in[17];
  const float* lin2_b   = (const float*)d_in[18];
  float* out = (float*)d_out;

  // Workspace layout (floats)
  float* ws     = (float*)d_ws;
  float* bufA   = ws;                         // xr1 (N*D) -> reused as xr2
  float* bufB   = ws + 8388608;               // acc1/h1  -> reused as acc2/h2
  float* xnew1  = ws + 16777216;              // N1*D
  float* xnew2  = ws + 20971520;              // N2*D
  int*   mapping = (int*)(ws + 23068672);     // NTOT ints
  float* gx1    = ws + 23134208;              // G*256
  float* gx2    = ws + 23166976;              // G*256

  // Layer 1: xr1 = x@w_rel1 ; acc1 = x@w_root1 + b_rel1   (bias folds here
  // since h = seg_sum(xr1[src]) + b_rel1 + x@w_root1)
  dual_gemm_kernel<IN_><<<NTOT / 32, 256, 0, stream>>>(
      x, w_rel1, w_root1, b_rel1, bufA, bufB);

  // h1 = relu(scatter-add over edges), whole graph accumulator in LDS (256KB)
  agg_relu_kernel<NPER_, false><<<G_, 512, NPER_ * D_ * 4, stream>>>(
      bufA, bufB, bufB, ei, nullptr);

  // TopK pool 1 (512 -> 256/graph); score_1 straight into d_out
  topk_pool_kernel<NPER_, K1_><<<G_, NPER_, 0, stream>>>(
      bufB, pool1_w, xnew1, mapping, out + OFF_SCORE);

  global_pool_kernel<<<G_, D_, 0, stream>>>(xnew1, gx1, K1_);

  // Layer 2 GEMMs on pooled nodes
  dual_gemm_kernel<D_><<<N1_ / 32, 256, 0, stream>>>(
      xnew1, w_rel2, w_root2, b_rel2, bufA, bufB);

  // h2 = relu(scatter-add over relabeled edges), 128KB LDS accumulator
  agg_relu_kernel<K1_, true><<<G_, 512, K1_ * D_ * 4, stream>>>(
      bufA, bufB, bufB, ei, mapping);

  // TopK pool 2 (256 -> 128/graph); no score output
  topk_pool_kernel<K1_, K2_><<<G_, K1_, 0, stream>>>(
      bufB, pool2_w, xnew2, mapping, nullptr);

  global_pool_kernel<<<G_, D_, 0, stream>>>(xnew2, gx2, K2_);

  // Head: g_emb, MLP, BN, log_softmax; also copies pool1_w to d_out
  head_kernel<<<G_, D_, 0, stream>>>(gx1, gx2, lin1_w, lin1_b,
                                     bn_gamma, bn_beta, bn_mean, bn_var,
                                     lin2_w, lin2_b, pool1_w, out);
}